// MnnCoreModule_22505628631233
// MI455X (gfx1250) — compile-verified
//
#include <hip/hip_runtime.h>
#include <math.h>
#include <stdint.h>

// ---------------------------------------------------------------------------
// MNN forward for MI455X (gfx1250).
// Tables (g, D1, D2) built on-device in fp64 (mirrors reference float64 build),
// stored interleaved [g, d1, d2, pad] in global scratch, then DMA'd into LDS
// once per workgroup with the CDNA5 Tensor Data Mover (tensor_load_to_lds +
// s_wait_tensorcnt). Streaming global traffic uses non-temporal hints so the
// 209KB LUT stays hot in the 192MB L2 while 335MB of one-touch data streams by.
// ---------------------------------------------------------------------------

#define TBL_N      13001              // grid points: linspace(-8, 5, 13001)
#define TBL_LAST   13000
#define TBL_PAD    13056              // padded entry count
#define TBL_STRIDE 4                  // g, d1, d2, pad  -> one b128 per entry
#define TBL_FLOATS (TBL_PAD * TBL_STRIDE)   // 52224 = 0xCC00 (< 65536, fits tile_dim0)
#define TBL_BYTES  (TBL_FLOATS * 4)         // 208896 B of LDS (<= 320KB/WGP)

typedef uint32_t u32x4 __attribute__((ext_vector_type(4)));
typedef uint32_t u32x8 __attribute__((ext_vector_type(8)));
typedef float    f32x4 __attribute__((ext_vector_type(4)));

// ---------------------------------------------------------------------------
// Table-build kernel: single workgroup, 512 threads, fp64.
//   g(x)  = sqrt(pi)/2 * exp(x^2) * erfc(-x)
//   D1    = cumtrapz(g)            - value at x=0 (index 8000)
//   q(x)  = pi/4 * exp(x^2) * erfc(-x)^2
//   h(x)  = exp(x^2) * cumtrapz(q)
//   D2    = cumtrapz(h)            - value at x=0
// ---------------------------------------------------------------------------

__device__ void block_cumtrapz(const double* __restrict__ f,
                               double* __restrict__ out,
                               double* part, double dx) {
    const int t  = threadIdx.x;
    const int C  = (TBL_N + 511) / 512;          // 26
    const int i0 = t * C;
    const int i1 = (i0 + C < TBL_N) ? (i0 + C) : TBL_N;
    double loc = 0.0;
    for (int i = (i0 > 1 ? i0 : 1); i < i1; ++i)
        loc += 0.5 * (f[i] + f[i - 1]) * dx;
    part[t] = loc;
    __syncthreads();
    if (t == 0) {
        double run = 0.0;
        for (int k = 0; k < 512; ++k) { double v = part[k]; part[k] = run; run += v; }
    }
    __syncthreads();
    double run = part[t];
    for (int i = i0; i < i1; ++i) {
        if (i >= 1) run += 0.5 * (f[i] + f[i - 1]) * dx;
        out[i] = run;                             // out[0] = 0
    }
    __syncthreads();
}

__global__ void __launch_bounds__(512)
build_tables(float* __restrict__ tab, double* __restrict__ A, double* __restrict__ B) {
    __shared__ double part[512];
    const int t  = threadIdx.x;
    const int C  = (TBL_N + 511) / 512;
    const int i0 = t * C;
    const int i1 = (i0 + C < TBL_N) ? (i0 + C) : TBL_N;
    const double dx  = 13.0 / 13000.0;            // 1e-3
    const double SP  = 0.8862269254527580;        // sqrt(pi)/2
    const double QPI = 0.7853981633974483;        // pi/4

    for (int i = i0; i < i1; ++i) {
        double x = -8.0 + dx * (double)i;
        double g = SP * exp(x * x) * erfc(-x);
        A[i] = g;
        tab[i * TBL_STRIDE + 0] = (float)g;
        tab[i * TBL_STRIDE + 3] = 0.0f;
    }
    __syncthreads();

    block_cumtrapz(A, B, part, dx);               // B = cumtrapz(g)
    double d1z = B[8000];                         // x = 0 exactly on grid
    for (int i = i0; i < i1; ++i)
        tab[i * TBL_STRIDE + 1] = (float)(B[i] - d1z);
    __syncthreads();

    for (int i = i0; i < i1; ++i) {
        double x = -8.0 + dx * (double)i;
        double e = erfc(-x);
        A[i] = QPI * exp(x * x) * e * e;          // q
    }
    __syncthreads();
    block_cumtrapz(A, B, part, dx);               // B = cumtrapz(q)
    for (int i = i0; i < i1; ++i) {
        double x = -8.0 + dx * (double)i;
        A[i] = exp(x * x) * B[i];                 // h
    }
    __syncthreads();
    block_cumtrapz(A, B, part, dx);               // B = cumtrapz(h) = D2 raw
    double d2z = B[8000];
    for (int i = i0; i < i1; ++i)
        tab[i * TBL_STRIDE + 2] = (float)(B[i] - d2z);

    for (int i = TBL_N + t; i < TBL_PAD; i += 512) {   // clear pad entries
        tab[i * TBL_STRIDE + 0] = 0.0f;
        tab[i * TBL_STRIDE + 1] = 0.0f;
        tab[i * TBL_STRIDE + 2] = 0.0f;
        tab[i * TBL_STRIDE + 3] = 0.0f;
    }
}

// ---------------------------------------------------------------------------
// TDM: stage the whole interleaved table (208896 B) into LDS with one DMA.
// D# built per CDNA5 ISA (1-row tile, 4-byte elements).
// ---------------------------------------------------------------------------
__device__ __forceinline__ void tdm_load_table(const float* gtab, uint32_t lds_off) {
    const uint32_t N = (uint32_t)TBL_FLOATS;      // 0xCC00 elements of 4 B
    uint64_t ga = (uint64_t)(uintptr_t)gtab;
    u32x4 g0;
    g0[0] = 1u;                                   // count=1, user descriptor
    g0[1] = lds_off;                              // lds_addr (bytes)
    g0[2] = (uint32_t)ga;                         // global_addr[31:0]
    g0[3] = (uint32_t)((ga >> 32) & 0x01FFFFFFu)  // global_addr[56:32]
          | 0x80000000u;                          // type = 2 ("image")
    u32x8 g1;
    g1[0] = 0x00020000u;                          // data_size = 2 (4 bytes)
    g1[1] = (N & 0xFFFFu) << 16;                  // tensor_dim0[15:0]
    g1[2] = (N >> 16) | (1u << 16);               // tensor_dim0[31:16] | tensor_dim1=1
    g1[3] = (N & 0xFFFFu) << 16;                  // tile_dim0 = N
    g1[4] = 1u;                                   // tile_dim1 = 1, tile_dim2 = 0
    g1[5] = N;                                    // tensor_dim0_stride[31:0]
    g1[6] = (N & 0xFFFFu) << 16;                  // tensor_dim1_stride[15:0]
    g1[7] = 0u;
    asm volatile("tensor_load_to_lds %0, %1" :: "s"(g0), "s"(g1) : "memory");
}

// ---------------------------------------------------------------------------
// Per-element math (predicated, mirrors the jnp.where reference).
// ---------------------------------------------------------------------------
struct MnnR { float ua, sa, chi; };

__device__ __forceinline__ MnnR mnn_one(const float* __restrict__ ltab, float u, float s) {
    const float SQRT_L = 0.22360679774997896f;    // sqrt(0.05)
    bool idx0 = (s <= 0.0f);
    bool idx1 = (s > 0.0f) && ((1.0f - u) < 10.0f * SQRT_L * s);
    bool idx2 = idx0 && (u > 1.0f);

    float s_safe = idx1 ? s : 1.0f;
    float inv = 1.0f / (SQRT_L * s_safe);
    float ub = (1.0f - u) * inv;
    float lb = (0.0f - u) * inv;

    // uniform-grid interp (clamped, like jnp.interp): idx = (z+8)*1000
    float tu = fminf(fmaxf(fmaf(ub, 1000.0f, 8000.0f), 0.0f), 13000.0f);
    int   iu = (int)tu; iu = (iu < TBL_LAST - 1) ? iu : (TBL_LAST - 1);
    float fu = tu - (float)iu;
    f32x4 a0 = *(const f32x4*)(ltab + iu * TBL_STRIDE);
    f32x4 a1 = *(const f32x4*)(ltab + iu * TBL_STRIDE + TBL_STRIDE);

    float tl = fminf(fmaxf(fmaf(lb, 1000.0f, 8000.0f), 0.0f), 13000.0f);
    int   il = (int)tl; il = (il < TBL_LAST - 1) ? il : (TBL_LAST - 1);
    float fl = tl - (float)il;
    f32x4 b0 = *(const f32x4*)(ltab + il * TBL_STRIDE);
    f32x4 b1 = *(const f32x4*)(ltab + il * TBL_STRIDE + TBL_STRIDE);

    float g_ub  = fmaf(fu, a1.x - a0.x, a0.x);
    float d1_ub = fmaf(fu, a1.y - a0.y, a0.y);
    float d2_ub = fmaf(fu, a1.z - a0.z, a0.z);
    float g_lb  = fmaf(fl, b1.x - b0.x, b0.x);
    float d1_lb = fmaf(fl, b1.y - b0.y, b0.y);
    float d2_lb = fmaf(fl, b1.z - b0.z, b0.z);

    float u_a1   = 40.0f * (d1_ub - d1_lb);                      // 2/L
    float u_safe = idx2 ? u : 2.0f;
    float isi2   = 5.0f - 20.0f * logf(1.0f - 1.0f / u_safe);    // T_REF - (1/L) log(...)
    float u_a    = idx1 ? u_a1 : (idx2 ? (1.0f / isi2) : 0.0f);

    float fano  = 3200.0f * (d2_ub - d2_lb) * u_a1 * u_a1;       // 8/L^2
    float s_pre = idx1 ? fano : ((idx0 && (u < 1.0f)) ? 1.0f : 0.0f);
    float prod  = s_pre * u_a;
    bool  pos   = prod > 0.0f;
    float s_a   = pos ? sqrtf(prod) : 0.0f;

    float s_a_safe = (idx1 && pos) ? s_a : 1.0f;
    float chi1 = (u_a1 * u_a1) / s_a_safe * (g_ub - g_lb) * 178.88543819998318f; // 2/L^1.5
    float chi2 = 6.324555320336759f * rsqrtf(isi2)                              // sqrt(2/L)
               * rsqrtf(idx2 ? (2.0f * u - 1.0f) : 1.0f);
    float chi  = idx1 ? chi1 : (idx2 ? chi2 : 0.0f);

    return {u_a, s_a, chi};
}

// ---------------------------------------------------------------------------
// Main kernel: TDM-stage LUT to LDS, then grid-stride float4 streaming with
// non-temporal global traffic (TH=NT) to keep the LUT resident in L2.
// ---------------------------------------------------------------------------
__global__ void __launch_bounds__(512)
mnn_main(const float* __restrict__ u, const float* __restrict__ s,
         const float* __restrict__ gtab, float* __restrict__ out, int n) {
    extern __shared__ float ltab[];

    // wave 0 issues the TDM DMA; uniform scalar branch so other waves truly skip it
    int wave = __builtin_amdgcn_readfirstlane((int)(threadIdx.x >> 5));
    if (wave == 0) {
        tdm_load_table(gtab, __builtin_amdgcn_groupstaticsize());
        __builtin_amdgcn_s_wait_tensorcnt(0);
    }
    __syncthreads();

    long long n4     = (n & 3) ? 0 : (n >> 2);
    long long gid    = (long long)blockIdx.x * blockDim.x + threadIdx.x;
    long long stride = (long long)gridDim.x * blockDim.x;

    const f32x4* u4 = (const f32x4*)u;
    const f32x4* s4 = (const f32x4*)s;
    f32x4* ou = (f32x4*)out;
    f32x4* os = (f32x4*)(out + n);
    f32x4* oc = (f32x4*)(out + 2LL * n);

    for (long long i = gid; i < n4; i += stride) {
        f32x4 uu = __builtin_nontemporal_load(u4 + i);
        f32x4 ss = __builtin_nontemporal_load(s4 + i);
        MnnR r0 = mnn_one(ltab, uu.x, ss.x);
        MnnR r1 = mnn_one(ltab, uu.y, ss.y);
        MnnR r2 = mnn_one(ltab, uu.z, ss.z);
        MnnR r3 = mnn_one(ltab, uu.w, ss.w);
        f32x4 vu = {r0.ua, r1.ua, r2.ua, r3.ua};
        f32x4 vs = {r0.sa, r1.sa, r2.sa, r3.sa};
        f32x4 vc = {r0.chi, r1.chi, r2.chi, r3.chi};
        __builtin_nontemporal_store(vu, ou + i);
        __builtin_nontemporal_store(vs, os + i);
        __builtin_nontemporal_store(vc, oc + i);
    }
    // scalar tail (also covers n not divisible by 4)
    for (long long i = 4 * n4 + gid; i < n; i += stride) {
        MnnR r = mnn_one(ltab, u[i], s[i]);
        out[i] = r.ua; out[n + i] = r.sa; out[2LL * n + i] = r.chi;
    }
}

// ---------------------------------------------------------------------------
extern "C" void kernel_launch(void* const* d_in, const int* in_sizes, int n_in,
                              void* d_out, int out_size, void* d_ws, size_t ws_size,
                              hipStream_t stream) {
    const float* u = (const float*)d_in[0];
    const float* s = (const float*)d_in[1];
    int n = in_sizes[0];

    // ws layout: [fp32 table: TBL_BYTES][fp64 A: TBL_PAD][fp64 B: TBL_PAD]
    float*  tab = (float*)d_ws;
    double* A   = (double*)((char*)d_ws + TBL_BYTES);
    double* B   = A + TBL_PAD;

    build_tables<<<1, 512, 0, stream>>>(tab, A, B);

    int blocks = 1024;                            // grid-stride; 208KB LDS -> 1 block/WGP
    mnn_main<<<blocks, 512, TBL_BYTES, stream>>>(u, s, tab, (float*)d_out, n);
}